// Cls_Loss_42331197670001
// MI455X (gfx1250) — compile-verified
//
#include <hip/hip_runtime.h>
#include <math.h>

typedef __attribute__((ext_vector_type(2))) float v2f;
typedef __attribute__((ext_vector_type(8))) float v8f;

#define A_DIM 512
#define C_NUM 345
#define N_SMP 128
#define M_BLK 64         // classes per workgroup = four 16-row WMMA tiles
#define N_TIL 4          // A tiles per workgroup
#define LDS_STRIDE 516   // 512 + 4 floats pad: conflict-free ds_load_b64 A fetch

// ---------------------------------------------------------------------------
// Kernel A: fused  sigma2[n,c] = dataW_nc * Sigma_{l(n)} * dataW_nc^T  and
//           dW_dMean[n,c]; writes aug_y[n,c] = y_s + 0.5*Lam*sig + Lam*dwm.
// grid = (6 class-blocks of 64, 128 samples), block = 128 threads (4 wave32)
// Each Sigma B-load pair feeds FOUR v_wmma_f32_16x16x4_f32 (four class tiles).
// ---------------------------------------------------------------------------
__global__ __launch_bounds__(128)
void aug_kernel(const float* __restrict__ W,        // [C,A]
                const float* __restrict__ y_s,      // [N,C]
                const int*   __restrict__ labels,   // [N]
                const float* __restrict__ lambda_p, // [1]
                const float* __restrict__ mean_s,   // [C,A]
                const float* __restrict__ mean_t,   // [C,A]
                const float* __restrict__ Sigma,    // [C,A,A]
                float* __restrict__ aug)            // [N,C]
{
    __shared__ float dataW[M_BLK * LDS_STRIDE];     // ~132 KB
    __shared__ float dmean[A_DIM];
    __shared__ float sig[M_BLK];
    __shared__ float dwm[M_BLK];

    const int n   = blockIdx.y;
    const int c0  = blockIdx.x * M_BLK;
    const int tid = threadIdx.x;
    const int lbl = labels[n];
    const float lam = *lambda_p;

    const float* Wl = W + (size_t)lbl * A_DIM;

    // Stage dataW block: dataW[r,k] = W[c0+r,k] - W[lbl,k] (clamped rows masked at output)
    for (int idx = tid; idx < M_BLK * A_DIM; idx += 128) {
        int r = idx >> 9;
        int k = idx & (A_DIM - 1);
        int c = c0 + r; if (c >= C_NUM) c = C_NUM - 1;
        dataW[r * LDS_STRIDE + k] = W[(size_t)c * A_DIM + k] - Wl[k];
    }
    for (int k = tid; k < A_DIM; k += 128)
        dmean[k] = mean_t[(size_t)lbl * A_DIM + k] - mean_s[(size_t)lbl * A_DIM + k];
    if (tid < M_BLK) { sig[tid] = 0.0f; dwm[tid] = 0.0f; }
    __syncthreads();

    const int lane = tid & 31;
    const int wave = tid >> 5;
    const int hi   = lane >> 4;   // K half-pair select
    const int lr   = lane & 15;   // M row (for A) / N col (for B)

    const float* Sg = Sigma + (size_t)lbl * A_DIM * A_DIM;

    // Each wave handles 8 of the 32 N-tiles; fuse diag(X * dataW^T) on the fly.
    v8f sacc[N_TIL] = {};
    for (int bt = wave; bt < 32; bt += 4) {
        const int col = bt * 16 + lr;
        v8f acc[N_TIL] = {};
        #pragma unroll 4
        for (int k = 0; k < A_DIM; k += 4) {
            // B (4x16 f32): VGPR0 = K even row, VGPR1 = K odd row, per lane half
            int krow = k + 2 * hi;
            v2f b;
            b.x = Sg[(size_t)krow * A_DIM + col];
            b.y = Sg[(size_t)(krow + 1) * A_DIM + col];
            // A tiles (16x4 f32): lanes0-15 K={k,k+1}, lanes16-31 K={k+2,k+3}
            #pragma unroll
            for (int t = 0; t < N_TIL; ++t) {
                v2f a = *(const v2f*)&dataW[(t * 16 + lr) * LDS_STRIDE + k + 2 * hi];
                acc[t] = __builtin_amdgcn_wmma_f32_16x16x4_f32(false, a, false, b,
                                                               (short)0, acc[t],
                                                               false, false);
            }
        }
        // D layout: VGPR r -> row r (lanes0-15) / r+8 (lanes16-31), lane%16 -> col
        #pragma unroll
        for (int t = 0; t < N_TIL; ++t) {
            #pragma unroll
            for (int r = 0; r < 8; ++r) {
                int rowg = r + 8 * hi;
                sacc[t][r] += acc[t][r] *
                    dataW[(t * 16 + rowg) * LDS_STRIDE + bt * 16 + lr];
            }
        }
    }

    // Reduce 16 column-partials per row. xor masks {1,2,4,8} never cross a
    // 16-lane half, so full-width shuffles are exact.
    #pragma unroll
    for (int t = 0; t < N_TIL; ++t) {
        #pragma unroll
        for (int r = 0; r < 8; ++r) {
            float v = sacc[t][r];
            v += __shfl_xor(v, 1);
            v += __shfl_xor(v, 2);
            v += __shfl_xor(v, 4);
            v += __shfl_xor(v, 8);
            if (lr == 0) atomicAdd(&sig[t * 16 + r + 8 * hi], v);   // ds_add_f32
        }
    }

    // dW_dMean: 2 threads per row (128 threads / 64 rows), strided dot with dmean
    {
        int row = tid >> 1;
        int sub = tid & 1;
        float p = 0.0f;
        for (int i = 0; i < A_DIM / 2; ++i) {
            int k = sub + 2 * i;
            p += dataW[row * LDS_STRIDE + k] * dmean[k];
        }
        atomicAdd(&dwm[row], p);
    }
    __syncthreads();

    if (tid < M_BLK) {
        int c = c0 + tid;
        if (c < C_NUM) {
            aug[(size_t)n * C_NUM + c] = y_s[(size_t)n * C_NUM + c]
                                       + 0.5f * lam * sig[tid] + lam * dwm[tid];
        }
    }
}

// ---------------------------------------------------------------------------
// Kernel B: per-sample log-softmax NLL
// ---------------------------------------------------------------------------
__global__ __launch_bounds__(128)
void nll_kernel(const float* __restrict__ aug, const int* __restrict__ labels,
                float* __restrict__ nll)
{
    __shared__ float red[128];
    const int n   = blockIdx.x;
    const int tid = threadIdx.x;
    const float* row = aug + (size_t)n * C_NUM;

    float m = -INFINITY;
    for (int c = tid; c < C_NUM; c += 128) m = fmaxf(m, row[c]);
    red[tid] = m; __syncthreads();
    for (int s = 64; s > 0; s >>= 1) {
        if (tid < s) red[tid] = fmaxf(red[tid], red[tid + s]);
        __syncthreads();
    }
    float mx = red[0];
    __syncthreads();

    float sum = 0.0f;
    for (int c = tid; c < C_NUM; c += 128) sum += expf(row[c] - mx);
    red[tid] = sum; __syncthreads();
    for (int s = 64; s > 0; s >>= 1) {
        if (tid < s) red[tid] += red[tid + s];
        __syncthreads();
    }
    if (tid == 0) {
        int lbl = labels[n];
        nll[n] = -(row[lbl] - mx - logf(red[0]));
    }
}

// ---------------------------------------------------------------------------
// Kernel C: deterministic mean over 128 samples
// ---------------------------------------------------------------------------
__global__ void mean_kernel(const float* __restrict__ nll, float* __restrict__ out)
{
    if (threadIdx.x == 0) {
        float s = 0.0f;
        for (int i = 0; i < N_SMP; ++i) s += nll[i];
        out[0] = s * (1.0f / (float)N_SMP);
    }
}

// ---------------------------------------------------------------------------
extern "C" void kernel_launch(void* const* d_in, const int* in_sizes, int n_in,
                              void* d_out, int out_size, void* d_ws, size_t ws_size,
                              hipStream_t stream)
{
    const float* fc_weight = (const float*)d_in[0];
    // d_in[1] = features_source (unused by the reference loss)
    const float* y_s       = (const float*)d_in[2];
    const int*   labels    = (const int*)  d_in[3];
    const float* Lambda    = (const float*)d_in[4];
    const float* mean_s    = (const float*)d_in[5];
    const float* mean_t    = (const float*)d_in[6];
    const float* Sigma     = (const float*)d_in[7];

    float* aug = (float*)d_ws;                 // [N, C]
    float* nll = aug + (size_t)N_SMP * C_NUM;  // [N]

    dim3 gridA((C_NUM + M_BLK - 1) / M_BLK, N_SMP);   // 6 x 128
    aug_kernel<<<gridA, 128, 0, stream>>>(fc_weight, y_s, labels, Lambda,
                                          mean_s, mean_t, Sigma, aug);
    nll_kernel<<<N_SMP, 128, 0, stream>>>(aug, labels, nll);
    mean_kernel<<<1, 32, 0, stream>>>(nll, (float*)d_out);
}